// DimeNetPlusPlusPotential_62603443306984
// MI455X (gfx1250) — compile-verified
//
#include <hip/hip_runtime.h>
#include <math.h>

namespace dnet {

constexpr int NE = 30720;      // edges
constexpr int NT = 430080;     // triplets
constexpr int NN = 2048;       // nodes
constexpr int NM = 128;        // molecules
constexpr float PI_F = 3.14159265358979323846f;
constexpr double PI_D = 3.14159265358979323846;

typedef __attribute__((ext_vector_type(2))) float v2f;
typedef __attribute__((ext_vector_type(8))) float v8f;

__device__ __forceinline__ float siluf(float x){ float s = 1.f/(1.f+expf(-x)); return x*s; }
__device__ __forceinline__ float silugf(float x){ float s = 1.f/(1.f+expf(-x)); return s*(1.f + x*(1.f-s)); }

// ======================================================================================
// FAST WMMA GEMM: requires M%64==0, N%16==0, K%4==0  (all hot GEMMs in this model).
// One wave computes a 64x16 tile: 4 accumulators share each B fragment (amortizes the
// strided B loads 4x), A/transposed-B fragments are 8-byte vector loads, no guards.
// ======================================================================================
__device__ __forceinline__ void epilogue16(v8f acc, float* __restrict__ C,
                                           float* __restrict__ preact, int row0, int N,
                                           int bcol, float bv, int act, int accum, int half)
{
#pragma unroll
  for (int r = 0; r < 8; ++r) {
    const int row = row0 + r + (half << 3);
    const size_t o = (size_t)row * N + bcol;
    float z = acc[r] + bv;
    if (preact) preact[o] = z;
    float y = (act == 1) ? siluf(z) : z;
    if (accum) C[o] += y; else C[o] = y;
  }
}

__global__ __launch_bounds__(32)
void gemm_wmma_fast(const float* __restrict__ A, const float* __restrict__ B,
                    const float* __restrict__ bias, float* __restrict__ C,
                    float* __restrict__ preact, int M, int N, int K,
                    int transB, int act, int accum)
{
  const int m0   = blockIdx.x << 6;   // 64 rows per wave
  const int n0   = blockIdx.y << 4;   // 16 cols
  const int lane = threadIdx.x & 31;
  const int half = lane >> 4;
  const int lr   = lane & 15;
  const int bcol = n0 + lr;
  const float* __restrict__ Ar = A + (size_t)(m0 + lr) * K;
  const size_t rowK = (size_t)16 * K;
  v8f c0 = {0.f,0.f,0.f,0.f,0.f,0.f,0.f,0.f};
  v8f c1 = c0, c2 = c0, c3 = c0;
  if (!transB) {
    const float* __restrict__ Bc = B + bcol;
#pragma unroll 4
    for (int k = 0; k < K; k += 4) {
      const int kk = k + (half << 1);
      v2f b; b.x = Bc[(size_t)kk * N]; b.y = Bc[(size_t)(kk + 1) * N];
      v2f a0 = *(const v2f*)(Ar + kk);
      v2f a1 = *(const v2f*)(Ar + rowK + kk);
      v2f a2 = *(const v2f*)(Ar + 2*rowK + kk);
      v2f a3 = *(const v2f*)(Ar + 3*rowK + kk);
      c0 = __builtin_amdgcn_wmma_f32_16x16x4_f32(false, a0, false, b, (short)0, c0, false, false);
      c1 = __builtin_amdgcn_wmma_f32_16x16x4_f32(false, a1, false, b, (short)0, c1, false, false);
      c2 = __builtin_amdgcn_wmma_f32_16x16x4_f32(false, a2, false, b, (short)0, c2, false, false);
      c3 = __builtin_amdgcn_wmma_f32_16x16x4_f32(false, a3, false, b, (short)0, c3, false, false);
    }
  } else {
    const float* __restrict__ Bc = B + (size_t)bcol * K;
#pragma unroll 4
    for (int k = 0; k < K; k += 4) {
      const int kk = k + (half << 1);
      v2f b  = *(const v2f*)(Bc + kk);
      v2f a0 = *(const v2f*)(Ar + kk);
      v2f a1 = *(const v2f*)(Ar + rowK + kk);
      v2f a2 = *(const v2f*)(Ar + 2*rowK + kk);
      v2f a3 = *(const v2f*)(Ar + 3*rowK + kk);
      c0 = __builtin_amdgcn_wmma_f32_16x16x4_f32(false, a0, false, b, (short)0, c0, false, false);
      c1 = __builtin_amdgcn_wmma_f32_16x16x4_f32(false, a1, false, b, (short)0, c1, false, false);
      c2 = __builtin_amdgcn_wmma_f32_16x16x4_f32(false, a2, false, b, (short)0, c2, false, false);
      c3 = __builtin_amdgcn_wmma_f32_16x16x4_f32(false, a3, false, b, (short)0, c3, false, false);
    }
  }
  const float bv = bias ? bias[bcol] : 0.f;
  epilogue16(c0, C, preact, m0,      N, bcol, bv, act, accum, half);
  epilogue16(c1, C, preact, m0 + 16, N, bcol, bv, act, accum, half);
  epilogue16(c2, C, preact, m0 + 32, N, bcol, bv, act, accum, half);
  epilogue16(c3, C, preact, m0 + 48, N, bcol, bv, act, accum, half);
}

// ============== generic guarded WMMA GEMM (only for tiny K=6/42 or N=6/8/42 cases) =====
__global__ __launch_bounds__(32)
void gemm_wmma_gen(const float* __restrict__ A, const float* __restrict__ B,
                   const float* __restrict__ bias, float* __restrict__ C,
                   float* __restrict__ preact, int M, int N, int K,
                   int transB, int act, int accum)
{
  const int m0   = blockIdx.x << 4;
  const int n0   = blockIdx.y << 4;
  const int lane = threadIdx.x & 31;
  const int half = lane >> 4;
  const int lr   = lane & 15;
  const int arow = m0 + lr;
  const int bcol = n0 + lr;
  const bool aok = arow < M;
  const bool bok = bcol < N;
  v8f acc = {0.f,0.f,0.f,0.f,0.f,0.f,0.f,0.f};
  for (int k = 0; k < K; k += 4) {
    const int kk = k + (half << 1);
    v2f a; a.x = 0.f; a.y = 0.f;
    if (aok) {
      if (kk     < K) a.x = A[(size_t)arow * K + kk];
      if (kk + 1 < K) a.y = A[(size_t)arow * K + kk + 1];
    }
    v2f b; b.x = 0.f; b.y = 0.f;
    if (bok) {
      if (!transB) {
        if (kk     < K) b.x = B[(size_t)kk       * N + bcol];
        if (kk + 1 < K) b.y = B[(size_t)(kk + 1) * N + bcol];
      } else {
        if (kk     < K) b.x = B[(size_t)bcol * K + kk];
        if (kk + 1 < K) b.y = B[(size_t)bcol * K + kk + 1];
      }
    }
    acc = __builtin_amdgcn_wmma_f32_16x16x4_f32(false, a, false, b, (short)0, acc, false, false);
  }
  if (!bok) return;
  const float bv = bias ? bias[bcol] : 0.f;
#pragma unroll
  for (int r = 0; r < 8; ++r) {
    const int row = m0 + r + (half << 3);
    if (row >= M) continue;
    const size_t o = (size_t)row * N + bcol;
    float z = acc[r] + bv;
    if (preact) preact[o] = z;
    float y = (act == 1) ? siluf(z) : z;
    if (accum) C[o] += y; else C[o] = y;
  }
}

// ================= elementwise =================
__global__ void k_fill(float* p, float v, long long n){ long long i=(long long)blockIdx.x*256+threadIdx.x; if(i<n) p[i]=v; }
__global__ void k_copy(float* d, const float* s, long long n){ long long i=(long long)blockIdx.x*256+threadIdx.x; if(i<n) d[i]=s[i]; }
__global__ void k_add (float* d, const float* s, long long n){ long long i=(long long)blockIdx.x*256+threadIdx.x; if(i<n) d[i]+=s[i]; }
__global__ void k_mul (float* d, const float* a, const float* b, long long n){ long long i=(long long)blockIdx.x*256+threadIdx.x; if(i<n) d[i]=a[i]*b[i]; }
__global__ void k_mul_add(float* d, const float* a, const float* b, long long n){ long long i=(long long)blockIdx.x*256+threadIdx.x; if(i<n) d[i]+=a[i]*b[i]; }
__global__ void k_silu_bwd(float* d, const float* g, const float* z, long long n){ long long i=(long long)blockIdx.x*256+threadIdx.x; if(i<n) d[i]=g[i]*silugf(z[i]); }
__global__ void k_mul_silu(float* d, const float* a, const float* z, long long n){ long long i=(long long)blockIdx.x*256+threadIdx.x; if(i<n) d[i]=a[i]*siluf(z[i]); }

// ================= constants: spherical Bessel zeros / norms / sph coef (fp64, matches reference) ========
__device__ double jl_d(int l, double x){
  double j0 = sin(x)/x;
  if (l==0) return j0;
  double jm=j0, jc = sin(x)/(x*x) - cos(x)/x;
  for (int ll=1; ll<l; ++ll){ double t=(2.0*ll+1.0)/x*jc - jm; jm=jc; jc=t; }
  return jc;
}
__global__ void k_init_consts(float* c){  // [0,42): zeros  [42,84): snorm  [84,91): sph
  if (threadIdx.x || blockIdx.x) return;
  double pts[13];
  for (int i=0;i<13;++i) pts[i]=(i+1)*PI_D;
  double zer[7][6];
  for (int r=0;r<6;++r) zer[0][r]=pts[r];
  int np = 13;
  for (int l=1;l<7;++l){
    double nw[12];
    for (int i=0;i<np-1;++i){
      double lo=pts[i], hi=pts[i+1];
      double flo=jl_d(l,lo);
      for (int it=0; it<100; ++it){
        double mid=0.5*(lo+hi);
        if (jl_d(l,mid)*flo > 0.0) lo=mid; else hi=mid;
      }
      nw[i]=0.5*(lo+hi);
    }
    np--;
    for (int i=0;i<np;++i) pts[i]=nw[i];
    for (int r=0;r<6;++r) zer[l][r]=pts[r];
  }
  for (int l=0;l<7;++l) for (int r=0;r<6;++r){
    c[l*6+r]    = (float)zer[l][r];
    c[42+l*6+r] = (float)(sqrt(2.0)/fabs(jl_d(l+1, zer[l][r])));
  }
  for (int l=0;l<7;++l) c[84+l] = (float)sqrt((2.0*l+1.0)/(4.0*PI_D));
}

// ================= edge geometry: rbf/srbf and d/dxs ==========================
__device__ __forceinline__ void jl_pair(int l, float t, float* jl, float* djl){
  float s = sinf(t), cc = cosf(t);
  float j0 = s/t;
  if (l==0){ *jl=j0; *djl = cc/t - s/(t*t); return; }
  float jm=j0, jc = s/(t*t) - cc/t;
  for (int ll=1; ll<l; ++ll){ float nx=(2.f*ll+1.f)/t*jc - jm; jm=jc; jc=nx; }
  *jl=jc; *djl = jm - (float)(l+1)/t*jc;
}
__global__ void k_edge_geom(const float* __restrict__ pos, const int* __restrict__ esrc,
                            const int* __restrict__ edst, const float* __restrict__ c,
                            float* dist, float* rbf, float* rbfd, float* srbf, float* srbfd)
{
  int e = blockIdx.x*256 + threadIdx.x;
  if (e >= NE) return;
  int s=esrc[e], d=edst[e];
  float rx=pos[3*d]-pos[3*s], ry=pos[3*d+1]-pos[3*s+1], rz=pos[3*d+2]-pos[3*s+2];
  float dd = sqrtf(rx*rx+ry*ry+rz*rz);
  dist[e]=dd;
  float x = dd*(1.f/5.f);
  float in = (x < 1.f) ? 1.f : 0.f;
  float x2=x*x, x4=x2*x2, x5=x4*x, x6=x5*x, x7=x6*x;
  float env  = (1.f/x - 28.f*x5 + 48.f*x6 - 21.f*x7)*in;           // p=6 envelope
  float denv = (-1.f/x2 - 140.f*x4 + 288.f*x5 - 147.f*x6)*in;
  for (int r=0;r<6;++r){
    float f=(r+1)*PI_F;
    float sn=sinf(f*x), cs=cosf(f*x);
    rbf [e*6+r]= env*sn;
    rbfd[e*6+r]= denv*sn + env*f*cs;
  }
  for (int l=0;l<7;++l) for (int r=0;r<6;++r){
    float z = c[l*6+r], nm = c[42+l*6+r];
    float jl, djl; jl_pair(l, x*z, &jl, &djl);
    srbf [e*42+l*6+r] = env*jl*nm;
    srbfd[e*42+l*6+r] = (denv*jl + env*djl*z)*nm;
  }
}

// ================= triplet geometry: cos(angle), sbf ==========================
__global__ void k_triplet(const float* __restrict__ pos, const int* __restrict__ esrc,
                          const int* __restrict__ edst, const int* __restrict__ ikj,
                          const int* __restrict__ iji, const float* __restrict__ srbf,
                          const float* __restrict__ c, float* cosv, float* sbf)
{
  int t = blockIdx.x*256 + threadIdx.x;
  if (t >= NT) return;
  int ej=iji[t], ek=ikj[t];
  int ti=edst[ej], tj=esrc[ej], tk=esrc[ek];
  float jx=pos[3*tj]-pos[3*ti], jy=pos[3*tj+1]-pos[3*ti+1], jz=pos[3*tj+2]-pos[3*ti+2];
  float kx=pos[3*tk]-pos[3*tj], ky=pos[3*tk+1]-pos[3*tj+1], kz=pos[3*tk+2]-pos[3*tj+2];
  float a = jx*kx+jy*ky+jz*kz;
  float r1 = sqrtf(jx*jx+jy*jy+jz*jz), r2 = sqrtf(kx*kx+ky*ky+kz*kz);
  float cv = a/(r1*r2);
  cosv[t]=cv;
  float P[7]; P[0]=1.f; P[1]=cv;
  for (int l=1;l<6;++l) P[l+1]=((2.f*l+1.f)*cv*P[l]-(float)l*P[l-1])/(float)(l+1);
  for (int l=0;l<7;++l){
    float cb = P[l]*c[84+l];
    for (int r=0;r<6;++r) sbf[(size_t)t*42+l*6+r] = srbf[(size_t)ek*42+l*6+r]*cb;
  }
}

// ================= graph gathers (all deterministic, no atomics) ==================
__global__ void k_concat(const float* __restrict__ embz, const int* __restrict__ z,
                         const int* __restrict__ esrc, const int* __restrict__ edst,
                         const float* __restrict__ yer, float* __restrict__ C)
{
  long long i=(long long)blockIdx.x*256+threadIdx.x;
  if (i >= (long long)NE*128) return;
  int e=(int)(i>>7), f=(int)(i&127);
  C[(size_t)e*384 + f]       = embz[(size_t)z[edst[e]]*128 + f];
  C[(size_t)e*384 + 128 + f] = embz[(size_t)z[esrc[e]]*128 + f];
  C[(size_t)e*384 + 256 + f] = yer[i];
}
// S[n,f] = sum over 15 incoming edges of te[e,f]
__global__ void k_node_sum(const float* __restrict__ te, float* __restrict__ S)
{
  long long i=(long long)blockIdx.x*256+threadIdx.x;
  if (i >= (long long)NN*128) return;
  int n=(int)(i>>7), f=(int)(i&127);
  int mol=n>>4, d=n&15;
  float acc=0.f;
  for (int s=0;s<16;++s){
    if (s==d) continue;
    int le = s*15 + (d - (d>s?1:0));
    acc += te[(size_t)(mol*240+le)*128 + f];
  }
  S[i]=acc;
}
__global__ void k_mol_sum(const float* __restrict__ Pn, float* __restrict__ g)
{
  long long i=(long long)blockIdx.x*256+threadIdx.x;
  if (i >= (long long)NM*128) return;
  int m=(int)(i>>7), f=(int)(i&127);
  float a=0.f;
  for (int k=0;k<16;++k) a += Pn[(size_t)(m*16+k)*128+f];
  g[i]=a;
}
__global__ void k_node_bcast(float* __restrict__ dPn, const float* __restrict__ dg)
{
  long long i=(long long)blockIdx.x*256+threadIdx.x;
  if (i >= (long long)NN*128) return;
  int n=(int)(i>>7), f=(int)(i&127);
  dPn[i] = dg[(size_t)(n>>4)*128+f];
}
__global__ void k_edge_gather(float* __restrict__ dt, const float* __restrict__ S,
                              const int* __restrict__ edst)
{
  long long i=(long long)blockIdx.x*256+threadIdx.x;
  if (i >= (long long)NE*128) return;
  int e=(int)(i>>7), f=(int)(i&127);
  dt[i] = S[(size_t)edst[e]*128+f];
}
// agg[e,f] = sum_{q<14} xkj2[ikj[e*14+q],f] * sb[e*14+q,f]   (idx_ji is contiguous)
__global__ void k_msg_agg(const float* __restrict__ xkj2, const float* __restrict__ sb,
                          const int* __restrict__ ikj, float* __restrict__ agg)
{
  long long i=(long long)blockIdx.x*256+threadIdx.x;
  if (i >= (long long)NE*64) return;
  int e=(int)(i>>6), f=(int)(i&63);
  float a=0.f;
  for (int q=0;q<14;++q){
    int t = e*14+q;
    a += xkj2[(size_t)ikj[t]*64+f]*sb[(size_t)t*64+f];
  }
  agg[i]=a;
}
__global__ void k_dsb(float* __restrict__ dsb, const float* __restrict__ dagg,
                      const float* __restrict__ xkj2, const int* __restrict__ ikj,
                      const int* __restrict__ iji)
{
  long long i=(long long)blockIdx.x*256+threadIdx.x;
  if (i >= (long long)NT*64) return;
  int t=(int)(i>>6), f=(int)(i&63);
  dsb[i] = dagg[(size_t)iji[t]*64+f]*xkj2[(size_t)ikj[t]*64+f];
}
// dxkj2[e,f] = sum over the 14 triplets with idx_kj==e of dagg[ji_edge,f]*sb[t,f]
__global__ void k_dxkj2(const float* __restrict__ dagg, const float* __restrict__ sb,
                        float* __restrict__ dxkj2)
{
  long long i=(long long)blockIdx.x*256+threadIdx.x;
  if (i >= (long long)NE*64) return;
  int e=(int)(i>>6), f=(int)(i&63);
  int mol=e/240, le=e%240, s=le/15, dd=le%15, d = dd + (dd>=s?1:0);
  float a=0.f;
  for (int iN=0;iN<16;++iN){
    if (iN==d || iN==s) continue;
    int eg  = mol*240 + d*15 + (iN - (iN>d?1:0));
    int off = s - (d<s?1:0) - (iN<s?1:0);
    int t   = eg*14 + off;
    a += dagg[(size_t)eg*64+f]*sb[(size_t)t*64+f];
  }
  dxkj2[i]=a;
}
// dsrbfA[e,q] += sum over 14 triplets with idx_kj==e of dsbf[t,q]*cbf(t, l=q/6)
__global__ void k_dsrbf(const float* __restrict__ dsbf, const float* __restrict__ cosv,
                        float* __restrict__ dsrbfA, const float* __restrict__ c)
{
  long long i=(long long)blockIdx.x*256+threadIdx.x;
  if (i >= (long long)NE*42) return;
  int e=(int)(i/42), q=(int)(i%42), l=q/6;
  int mol=e/240, le=e%240, s=le/15, dd=le%15, d = dd + (dd>=s?1:0);
  float sph = c[84+l];
  float acc=0.f;
  for (int iN=0;iN<16;++iN){
    if (iN==d || iN==s) continue;
    int eg  = mol*240 + d*15 + (iN - (iN>d?1:0));
    int off = s - (d<s?1:0) - (iN<s?1:0);
    int t   = eg*14 + off;
    float cv = cosv[t];
    float Pl;
    if (l==0) Pl=1.f;
    else if (l==1) Pl=cv;
    else { float pm=1.f, pc=cv;
      for (int u=1;u<l;++u){ float pn=((2.f*u+1.f)*cv*pc-(float)u*pm)/(float)(u+1); pm=pc; pc=pn; }
      Pl=pc; }
    acc += dsbf[(size_t)t*42+q]*(Pl*sph);
  }
  dsrbfA[i] += acc;
}
__global__ void k_dcos(const float* __restrict__ dsbf, const float* __restrict__ srbf,
                       const int* __restrict__ ikj, const float* __restrict__ cosv,
                       float* __restrict__ dcosA, const float* __restrict__ c)
{
  int t = blockIdx.x*256 + threadIdx.x;
  if (t >= NT) return;
  int ek = ikj[t];
  float cv = cosv[t];
  float P[7], dP[7];
  P[0]=1.f; dP[0]=0.f; P[1]=cv; dP[1]=1.f;
  for (int l=1;l<6;++l){
    P [l+1]=((2.f*l+1.f)*cv*P[l]-(float)l*P[l-1])/(float)(l+1);
    dP[l+1]=((2.f*l+1.f)*(P[l]+cv*dP[l])-(float)l*dP[l-1])/(float)(l+1);
  }
  float acc=0.f;
  for (int l=0;l<7;++l){
    float s=0.f;
    for (int r=0;r<6;++r) s += dsbf[(size_t)t*42+l*6+r]*srbf[(size_t)ek*42+l*6+r];
    acc += s*c[84+l]*dP[l];
  }
  dcosA[t] += acc;
}
__global__ void k_xs_grad(const float* __restrict__ drbfA, const float* __restrict__ rbfd,
                          const float* __restrict__ dsrbfA, const float* __restrict__ srbfd,
                          float* __restrict__ gdist)
{
  int e = blockIdx.x*256 + threadIdx.x;
  if (e >= NE) return;
  float a=0.f;
  for (int r=0;r<6;++r)  a += drbfA[(size_t)e*6+r]*rbfd[(size_t)e*6+r];
  for (int q=0;q<42;++q) a += dsrbfA[(size_t)e*42+q]*srbfd[(size_t)e*42+q];
  gdist[e] = a*(1.f/5.f);
}

// ================= forces: per-node deterministic gather of all contributions ===========
__device__ __forceinline__ void trip_grad(int t, int role, float dc,
    const float* __restrict__ pos, const int* __restrict__ esrc, const int* __restrict__ edst,
    const int* __restrict__ ikj, const int* __restrict__ iji, float* gx, float* gy, float* gz)
{
  int ej=iji[t], ek=ikj[t];
  int ti=edst[ej], tj=esrc[ej], tk=esrc[ek];
  float jx=pos[3*tj]-pos[3*ti], jy=pos[3*tj+1]-pos[3*ti+1], jz=pos[3*tj+2]-pos[3*ti+2];
  float kx=pos[3*tk]-pos[3*tj], ky=pos[3*tk+1]-pos[3*tj+1], kz=pos[3*tk+2]-pos[3*tj+2];
  float a = jx*kx+jy*ky+jz*kz;
  float r1s=jx*jx+jy*jy+jz*jz, r2s=kx*kx+ky*ky+kz*kz;
  float inv = 1.f/(sqrtf(r1s)*sqrtf(r2s));
  float c1=a/r1s, c2=a/r2s;
  float gjx=(kx-c1*jx)*inv, gjy=(ky-c1*jy)*inv, gjz=(kz-c1*jz)*inv;  // dcos/dpji
  float gkx=(jx-c2*kx)*inv, gky=(jy-c2*ky)*inv, gkz=(jz-c2*kz)*inv;  // dcos/dpkj
  if (role==0){ *gx -= gjx*dc; *gy -= gjy*dc; *gz -= gjz*dc; }                         // node = t_i
  else if (role==1){ *gx += (gjx-gkx)*dc; *gy += (gjy-gky)*dc; *gz += (gjz-gkz)*dc; }  // node = t_j
  else { *gx += gkx*dc; *gy += gky*dc; *gz += gkz*dc; }                                // node = t_k
}
__global__ void k_forces(const float* __restrict__ pos, const int* __restrict__ esrc,
                         const int* __restrict__ edst, const int* __restrict__ ikj,
                         const int* __restrict__ iji, const float* __restrict__ gdist,
                         const float* __restrict__ dcos, float* __restrict__ out)
{
  int n = blockIdx.x*256 + threadIdx.x;
  if (n >= NN) return;
  int mol=n>>4, ln=n&15, nb=mol<<4;
  float px=pos[3*n], py=pos[3*n+1], pz=pos[3*n+2];
  float gx=0.f, gy=0.f, gz=0.f;
  // distance-path gradients (30 incident edges)
  for (int d=0; d<16; ++d){
    if (d==ln) continue;
    float qx=pos[3*(nb+d)], qy=pos[3*(nb+d)+1], qz=pos[3*(nb+d)+2];
    float rx=qx-px, ry=qy-py, rz=qz-pz;
    float dd = sqrtf(rx*rx+ry*ry+rz*rz);
    {
      int e = mol*240 + ln*15 + (d - (d>ln?1:0));   // outgoing (n->d): ddist/dp_n = -rel/dist
      float s = gdist[e]/dd;
      gx -= s*rx; gy -= s*ry; gz -= s*rz;
    }
    {
      int e = mol*240 + d*15 + (ln - (ln>d?1:0));   // incoming (d->n): ddist/dp_n = +rel/dist
      float s = gdist[e]/dd;
      gx += s*rx; gy += s*ry; gz += s*rz;
    }
  }
  // role i: ji edges with dst == n
  for (int sp=0; sp<16; ++sp){
    if (sp==ln) continue;
    int eg = mol*240 + sp*15 + (ln - (ln>sp?1:0));
    for (int q=0;q<14;++q){ int t=eg*14+q; trip_grad(t,0,dcos[t],pos,esrc,edst,ikj,iji,&gx,&gy,&gz); }
  }
  // role j: ji edges with src == n
  for (int d=0; d<16; ++d){
    if (d==ln) continue;
    int eg = mol*240 + ln*15 + (d - (d>ln?1:0));
    for (int q=0;q<14;++q){ int t=eg*14+q; trip_grad(t,1,dcos[t],pos,esrc,edst,ikj,iji,&gx,&gy,&gz); }
  }
  // role k: ji edges (sp->dp) with sp,dp != n
  for (int sp=0; sp<16; ++sp){
    if (sp==ln) continue;
    for (int dp=0; dp<16; ++dp){
      if (dp==sp || dp==ln) continue;
      int eg  = mol*240 + sp*15 + (dp - (dp>sp?1:0));
      int off = ln - (sp<ln?1:0) - (dp<ln?1:0);
      int t   = eg*14 + off;
      trip_grad(t,2,dcos[t],pos,esrc,edst,ikj,iji,&gx,&gy,&gz);
    }
  }
  out[3*n]=-gx; out[3*n+1]=-gy; out[3*n+2]=-gz;
}

// ================= head small kernels =================
__global__ void k_head_last(const float* __restrict__ h, const float* __restrict__ w,
                            const float* __restrict__ b, float* __restrict__ out)
{
  int m = blockIdx.x*64 + threadIdx.x;
  if (m >= NM) return;
  float a = b[0];
  for (int j=0;j<64;++j) a += h[(size_t)m*64+j]*w[j];
  out[m]=a;
}
__global__ void k_head_binit(float* __restrict__ dh2, const float* __restrict__ w)
{
  int i = blockIdx.x*256 + threadIdx.x;
  if (i >= NM*64) return;
  dh2[i] = w[i & 63];
}

// ================= host-side orchestration =================
struct Bufs {
  hipStream_t st;
  const float* pos; const int *z,*esrc,*edst,*ikj,*iji;
  const float *embz,*er_w,*er_b,*el_w,*el_b;
  struct IB { const float *jiw,*jib,*kjw,*kjb,*r1,*r2,*s1,*s2,*dn,*up,*lw,*lb,
              *b1w,*b1b,*b2w,*b2b,*a1w,*a1b,*a2w,*a2b,*a3w,*a3b,*a4w,*a4b; } ib[4];
  struct OB { const float *lrbf,*lup,*w0,*b0,*w1,*b1,*w2,*b2,*lout; } ob[5];
  const float *h0w,*h0b,*h1w,*h1b,*h2w,*h2b,*h3w,*h3b;
  float *consts,*dist,*rbf,*rbfd,*srbf,*srbfd,*cosv,*sbf;
  float *Zer,*Yer,*Zemb,*C384;
  float *x[5], *Pn, *Zo;
  float *gmol,*Zh0,*y0,*Zh1,*y1,*Zh2,*y2,*dh2s,*th0,*th1,*dgm;
  float *dPn,*drbfA,*dsrbfA,*dcosA,*gdist,*dX,*dXn;
  float *Zji,*Zkj,*g2,*Zup,*Zb1,*Zb2,*Zlin,*Za1,*Za2,*Za3,*Za4;
  float *v0,*v1,*v2,*hA,*hB,*dh;
  float *Zdn,*xkj2,*agg,*dagg,*dxkj2,*e64t;
  float *t8,*psiT,*sb,*dsb;
  float *u0,*u1,*nbuf;
};

static inline dim3 gr1(long long n){ return dim3((unsigned)((n+255)/256)); }
static void ggemm(Bufs& B, const float* A, const float* W, const float* bias,
                  float* C, float* pre, int M, int N, int K, int tB, int act, int acc)
{
  if (((M & 63) == 0) && ((N & 15) == 0) && ((K & 3) == 0)) {
    dim3 g(M >> 6, N >> 4);
    gemm_wmma_fast<<<g, 32, 0, B.st>>>(A, W, bias, C, pre, M, N, K, tB, act, acc);
  } else {
    dim3 g((M + 15) / 16, (N + 15) / 16);
    gemm_wmma_gen<<<g, 32, 0, B.st>>>(A, W, bias, C, pre, M, N, K, tB, act, acc);
  }
}

static void out_fwd(Bufs& B, int b, const float* xb)
{
  float* Zo0 = B.Zo + (size_t)(b*3+0)*NN*256;
  float* Zo1 = B.Zo + (size_t)(b*3+1)*NN*256;
  float* Zo2 = B.Zo + (size_t)(b*3+2)*NN*256;
  ggemm(B, B.rbf, B.ob[b].lrbf, nullptr, B.v2, nullptr, NE, 128, 6, 0, 0, 0);   // rbfW
  k_mul<<<gr1((long long)NE*128),256,0,B.st>>>(B.v1, B.v2, xb, (long long)NE*128);
  k_node_sum<<<gr1((long long)NN*128),256,0,B.st>>>(B.v1, B.nbuf);
  ggemm(B, B.nbuf, B.ob[b].lup, nullptr, B.u0, nullptr, NN, 256, 128, 0, 0, 0);
  ggemm(B, B.u0, B.ob[b].w0, B.ob[b].b0, B.u1, Zo0, NN, 256, 256, 0, 1, 0);
  ggemm(B, B.u1, B.ob[b].w1, B.ob[b].b1, B.u0, Zo1, NN, 256, 256, 0, 1, 0);
  ggemm(B, B.u0, B.ob[b].w2, B.ob[b].b2, B.u1, Zo2, NN, 256, 256, 0, 1, 0);
  ggemm(B, B.u1, B.ob[b].lout, nullptr, B.Pn, nullptr, NN, 128, 256, 0, 0, 1);  // Pn +=
}

static void out_bwd(Bufs& B, int b, const float* xb, float* dXtarget)
{
  float* Zo0 = B.Zo + (size_t)(b*3+0)*NN*256;
  float* Zo1 = B.Zo + (size_t)(b*3+1)*NN*256;
  float* Zo2 = B.Zo + (size_t)(b*3+2)*NN*256;
  long long n256 = (long long)NN*256, nE = (long long)NE*128;
  ggemm(B, B.dPn, B.ob[b].lout, nullptr, B.u0, nullptr, NN, 256, 128, 1, 0, 0);
  k_silu_bwd<<<gr1(n256),256,0,B.st>>>(B.u0, B.u0, Zo2, n256);
  ggemm(B, B.u0, B.ob[b].w2, nullptr, B.u1, nullptr, NN, 256, 256, 1, 0, 0);
  k_silu_bwd<<<gr1(n256),256,0,B.st>>>(B.u1, B.u1, Zo1, n256);
  ggemm(B, B.u1, B.ob[b].w1, nullptr, B.u0, nullptr, NN, 256, 256, 1, 0, 0);
  k_silu_bwd<<<gr1(n256),256,0,B.st>>>(B.u0, B.u0, Zo0, n256);
  ggemm(B, B.u0, B.ob[b].w0, nullptr, B.u1, nullptr, NN, 256, 256, 1, 0, 0);
  ggemm(B, B.u1, B.ob[b].lup, nullptr, B.nbuf, nullptr, NN, 128, 256, 1, 0, 0);  // dS
  k_edge_gather<<<gr1(nE),256,0,B.st>>>(B.v1, B.nbuf, B.edst);                   // dt_e
  ggemm(B, B.rbf, B.ob[b].lrbf, nullptr, B.v2, nullptr, NE, 128, 6, 0, 0, 0);    // rbfW
  k_mul_add<<<gr1(nE),256,0,B.st>>>(dXtarget, B.v1, B.v2, nE);
  k_mul<<<gr1(nE),256,0,B.st>>>(B.hA, B.v1, xb, nE);
  ggemm(B, B.hA, B.ob[b].lrbf, nullptr, B.drbfA, nullptr, NE, 6, 128, 1, 0, 1);
}

static void int_fwd(Bufs& B, int b, const float* xb, float* xout)
{
  const Bufs::IB& ib = B.ib[b];
  long long nE = (long long)NE*128;
  ggemm(B, xb, ib.jiw, ib.jib, B.v0, B.Zji, NE, 128, 128, 0, 1, 0);     // v0 = x_ji
  ggemm(B, xb, ib.kjw, ib.kjb, B.v1, B.Zkj, NE, 128, 128, 0, 1, 0);     // v1 = silu(Zkj)
  ggemm(B, B.rbf, ib.r1, nullptr, B.t8, nullptr, NE, 8, 6, 0, 0, 0);
  ggemm(B, B.t8, ib.r2, nullptr, B.g2, nullptr, NE, 128, 8, 0, 0, 0);
  k_mul<<<gr1(nE),256,0,B.st>>>(B.v2, B.v1, B.g2, nE);                  // xkj1
  ggemm(B, B.v2, ib.dn, nullptr, B.xkj2, B.Zdn, NE, 64, 128, 0, 1, 0);
  ggemm(B, B.sbf, ib.s1, nullptr, B.psiT, nullptr, NT, 8, 42, 0, 0, 0);
  ggemm(B, B.psiT, ib.s2, nullptr, B.sb, nullptr, NT, 64, 8, 0, 0, 0);
  k_msg_agg<<<gr1((long long)NE*64),256,0,B.st>>>(B.xkj2, B.sb, B.ikj, B.agg);
  k_copy<<<gr1(nE),256,0,B.st>>>(B.hA, B.v0, nE);
  ggemm(B, B.agg, ib.up, nullptr, B.hA, B.Zup, NE, 128, 64, 0, 1, 1);    // hA = h1
  ggemm(B, B.hA, ib.b1w, ib.b1b, B.v1, B.Zb1, NE, 128, 128, 0, 1, 0);
  ggemm(B, B.v1, ib.b2w, ib.b2b, B.hA, B.Zb2, NE, 128, 128, 0, 1, 1);   // hA = h2
  k_copy<<<gr1(nE),256,0,B.st>>>(B.hB, xb, nE);
  ggemm(B, B.hA, ib.lw, ib.lb, B.hB, B.Zlin, NE, 128, 128, 0, 1, 1);    // hB = h3
  ggemm(B, B.hB, ib.a1w, ib.a1b, B.v1, B.Za1, NE, 128, 128, 0, 1, 0);
  ggemm(B, B.v1, ib.a2w, ib.a2b, B.hB, B.Za2, NE, 128, 128, 0, 1, 1);   // hB = h4
  ggemm(B, B.hB, ib.a3w, ib.a3b, B.v1, B.Za3, NE, 128, 128, 0, 1, 0);
  k_copy<<<gr1(nE),256,0,B.st>>>(xout, B.hB, nE);
  ggemm(B, B.v1, ib.a4w, ib.a4b, xout, B.Za4, NE, 128, 128, 0, 1, 1);   // xout = x_{b+1}
}

static void int_bwd(Bufs& B, int b, const float* xb, float* dX, float* dXn)
{
  const Bufs::IB& ib = B.ib[b];
  long long nE = (long long)NE*128, n64 = (long long)NE*64, nT64 = (long long)NT*64;
  int_fwd(B, b, xb, B.dh);                 // recompute preacts/sb/xkj2 (dh used as throwaway xout)
  // after[1]
  k_silu_bwd<<<gr1(nE),256,0,B.st>>>(B.v0, dX, B.Za4, nE);
  ggemm(B, B.v0, ib.a4w, nullptr, B.v1, nullptr, NE, 128, 128, 1, 0, 0);
  k_silu_bwd<<<gr1(nE),256,0,B.st>>>(B.v1, B.v1, B.Za3, nE);
  k_copy<<<gr1(nE),256,0,B.st>>>(B.dh, dX, nE);
  ggemm(B, B.v1, ib.a3w, nullptr, B.dh, nullptr, NE, 128, 128, 1, 0, 1);  // dh = dh4
  // after[0]
  k_silu_bwd<<<gr1(nE),256,0,B.st>>>(B.v0, B.dh, B.Za2, nE);
  ggemm(B, B.v0, ib.a2w, nullptr, B.v1, nullptr, NE, 128, 128, 1, 0, 0);
  k_silu_bwd<<<gr1(nE),256,0,B.st>>>(B.v1, B.v1, B.Za1, nE);
  ggemm(B, B.v1, ib.a1w, nullptr, B.dh, nullptr, NE, 128, 128, 1, 0, 1);  // dh = dh3
  // h3 = silu(Zlin) + x
  k_add<<<gr1(nE),256,0,B.st>>>(dXn, B.dh, nE);
  k_silu_bwd<<<gr1(nE),256,0,B.st>>>(B.v0, B.dh, B.Zlin, nE);
  ggemm(B, B.v0, ib.lw, nullptr, B.dh, nullptr, NE, 128, 128, 1, 0, 0);   // dh = dh2
  // before[0]
  k_silu_bwd<<<gr1(nE),256,0,B.st>>>(B.v0, B.dh, B.Zb2, nE);
  ggemm(B, B.v0, ib.b2w, nullptr, B.v1, nullptr, NE, 128, 128, 1, 0, 0);
  k_silu_bwd<<<gr1(nE),256,0,B.st>>>(B.v1, B.v1, B.Zb1, nE);
  ggemm(B, B.v1, ib.b1w, nullptr, B.dh, nullptr, NE, 128, 128, 1, 0, 1);  // dh = dh1
  // x_ji path
  k_silu_bwd<<<gr1(nE),256,0,B.st>>>(B.v0, B.dh, B.Zji, nE);
  ggemm(B, B.v0, ib.jiw, nullptr, dXn, nullptr, NE, 128, 128, 1, 0, 1);
  // message path
  k_silu_bwd<<<gr1(nE),256,0,B.st>>>(B.v0, B.dh, B.Zup, nE);
  ggemm(B, B.v0, ib.up, nullptr, B.dagg, nullptr, NE, 64, 128, 1, 0, 0);
  k_dsb<<<gr1(nT64),256,0,B.st>>>(B.dsb, B.dagg, B.xkj2, B.ikj, B.iji);
  k_dxkj2<<<gr1(n64),256,0,B.st>>>(B.dagg, B.sb, B.dxkj2);
  ggemm(B, B.dsb, ib.s2, nullptr, B.psiT, nullptr, NT, 8, 64, 1, 0, 0);
  ggemm(B, B.psiT, ib.s1, nullptr, B.sb, nullptr, NT, 42, 8, 1, 0, 0);    // sb := dsbf (sb dead)
  k_dsrbf<<<gr1((long long)NE*42),256,0,B.st>>>(B.sb, B.cosv, B.dsrbfA, B.consts);
  k_dcos<<<gr1((long long)NT),256,0,B.st>>>(B.sb, B.srbf, B.ikj, B.cosv, B.dcosA, B.consts);
  // xkj2 -> xkj1 -> (skj, g2) paths
  k_silu_bwd<<<gr1(n64),256,0,B.st>>>(B.e64t, B.dxkj2, B.Zdn, n64);
  ggemm(B, B.e64t, ib.dn, nullptr, B.v0, nullptr, NE, 128, 64, 1, 0, 0);  // dxkj1
  k_mul<<<gr1(nE),256,0,B.st>>>(B.v1, B.v0, B.g2, nE);                    // dskj
  k_mul_silu<<<gr1(nE),256,0,B.st>>>(B.v2, B.v0, B.Zkj, nE);              // dg2
  k_silu_bwd<<<gr1(nE),256,0,B.st>>>(B.v1, B.v1, B.Zkj, nE);              // dZkj
  ggemm(B, B.v1, ib.kjw, nullptr, dXn, nullptr, NE, 128, 128, 1, 0, 1);
  ggemm(B, B.v2, ib.r2, nullptr, B.t8, nullptr, NE, 8, 128, 1, 0, 0);
  ggemm(B, B.t8, ib.r1, nullptr, B.drbfA, nullptr, NE, 6, 8, 1, 0, 1);
}

struct Alloc {
  char* base; size_t off;
  float* get(size_t nfl){ size_t b=off; off += ((nfl*sizeof(float)+255)/256)*256; return (float*)(base+b); }
};

} // namespace dnet

extern "C" void kernel_launch(void* const* d_in, const int* in_sizes, int n_in,
                              void* d_out, int out_size, void* d_ws, size_t ws_size,
                              hipStream_t stream) {
  using namespace dnet;
  (void)in_sizes; (void)n_in; (void)out_size; (void)ws_size;
  Bufs B;
  B.st = stream;
  B.pos  = (const float*)d_in[0];
  B.z    = (const int*)d_in[1];
  B.esrc = (const int*)d_in[3];
  B.edst = (const int*)d_in[4];
  B.ikj  = (const int*)d_in[5];
  B.iji  = (const int*)d_in[6];
  int c = 7;
  auto nf = [&]() { return (const float*)d_in[c++]; };
  nf();                       // freq (recomputed on device)
  B.embz = nf();
  B.er_w = nf(); B.er_b = nf();
  B.el_w = nf(); B.el_b = nf();
  for (int b=0;b<4;++b){
    Bufs::IB& ib = B.ib[b];
    ib.jiw=nf(); ib.jib=nf(); ib.kjw=nf(); ib.kjb=nf();
    ib.r1=nf(); ib.r2=nf(); ib.s1=nf(); ib.s2=nf();
    ib.dn=nf(); ib.up=nf(); ib.lw=nf(); ib.lb=nf();
    ib.b1w=nf(); ib.b1b=nf(); ib.b2w=nf(); ib.b2b=nf();
    ib.a1w=nf(); ib.a1b=nf(); ib.a2w=nf(); ib.a2b=nf();
    ib.a3w=nf(); ib.a3b=nf(); ib.a4w=nf(); ib.a4b=nf();
  }
  for (int b=0;b<5;++b){
    Bufs::OB& ob = B.ob[b];
    ob.lrbf=nf(); ob.lup=nf();
    ob.w0=nf(); ob.b0=nf(); ob.w1=nf(); ob.b1=nf(); ob.w2=nf(); ob.b2=nf();
    ob.lout=nf();
  }
  B.h0w=nf(); B.h0b=nf(); B.h1w=nf(); B.h1b=nf();
  B.h2w=nf(); B.h2b=nf(); B.h3w=nf(); B.h3b=nf();

  // ---------------- workspace layout ----------------
  Alloc al{(char*)d_ws, 0};
  B.consts = al.get(128);
  B.dist = al.get(NE);
  B.rbf  = al.get((size_t)NE*6);  B.rbfd  = al.get((size_t)NE*6);
  B.srbf = al.get((size_t)NE*42); B.srbfd = al.get((size_t)NE*42);
  B.cosv = al.get(NT);
  B.sbf  = al.get((size_t)NT*42);
  B.Zer  = al.get((size_t)NE*128); B.Yer = al.get((size_t)NE*128);
  B.Zemb = al.get((size_t)NE*128); B.C384 = al.get((size_t)NE*384);
  for (int i=0;i<5;++i) B.x[i] = al.get((size_t)NE*128);
  B.Pn = al.get((size_t)NN*128);
  B.Zo = al.get((size_t)NN*256*15);
  B.gmol=al.get(NM*128); B.Zh0=al.get(NM*128); B.y0=al.get(NM*128);
  B.Zh1=al.get(NM*64); B.y1=al.get(NM*64); B.Zh2=al.get(NM*64); B.y2=al.get(NM*64);
  B.dh2s=al.get(NM*64); B.th0=al.get(NM*128); B.th1=al.get(NM*64); B.dgm=al.get(NM*128);
  B.dPn=al.get((size_t)NN*128);
  B.drbfA=al.get((size_t)NE*6); B.dsrbfA=al.get((size_t)NE*42);
  B.dcosA=al.get(NT); B.gdist=al.get(NE);
  B.dX=al.get((size_t)NE*128); B.dXn=al.get((size_t)NE*128);
  B.Zji=al.get((size_t)NE*128); B.Zkj=al.get((size_t)NE*128); B.g2=al.get((size_t)NE*128);
  B.Zup=al.get((size_t)NE*128); B.Zb1=al.get((size_t)NE*128); B.Zb2=al.get((size_t)NE*128);
  B.Zlin=al.get((size_t)NE*128);
  B.Za1=al.get((size_t)NE*128); B.Za2=al.get((size_t)NE*128);
  B.Za3=al.get((size_t)NE*128); B.Za4=al.get((size_t)NE*128);
  B.v0=al.get((size_t)NE*128); B.v1=al.get((size_t)NE*128); B.v2=al.get((size_t)NE*128);
  B.hA=al.get((size_t)NE*128); B.hB=al.get((size_t)NE*128); B.dh=al.get((size_t)NE*128);
  B.Zdn=al.get((size_t)NE*64); B.xkj2=al.get((size_t)NE*64); B.agg=al.get((size_t)NE*64);
  B.dagg=al.get((size_t)NE*64); B.dxkj2=al.get((size_t)NE*64); B.e64t=al.get((size_t)NE*64);
  B.t8=al.get((size_t)NE*8);
  B.psiT=al.get((size_t)NT*8);
  B.sb=al.get((size_t)NT*64); B.dsb=al.get((size_t)NT*64);
  B.u0=al.get((size_t)NN*256); B.u1=al.get((size_t)NN*256); B.nbuf=al.get((size_t)NN*128);

  float* preds  = (float*)d_out;
  float* forces = (float*)d_out + NM;
  long long nE = (long long)NE*128;

  // ---------------- forward ----------------
  k_init_consts<<<1,32,0,stream>>>(B.consts);
  k_edge_geom<<<gr1(NE),256,0,stream>>>(B.pos,B.esrc,B.edst,B.consts,B.dist,B.rbf,B.rbfd,B.srbf,B.srbfd);
  k_triplet<<<gr1(NT),256,0,stream>>>(B.pos,B.esrc,B.edst,B.ikj,B.iji,B.srbf,B.consts,B.cosv,B.sbf);

  ggemm(B, B.rbf, B.er_w, B.er_b, B.Yer, B.Zer, NE, 128, 6, 0, 1, 0);
  k_concat<<<gr1(nE),256,0,stream>>>(B.embz,B.z,B.esrc,B.edst,B.Yer,B.C384);
  ggemm(B, B.C384, B.el_w, B.el_b, B.x[0], B.Zemb, NE, 128, 384, 0, 1, 0);

  k_fill<<<gr1((long long)NN*128),256,0,stream>>>(B.Pn, 0.f, (long long)NN*128);
  out_fwd(B, 0, B.x[0]);
  for (int b=0;b<4;++b){
    int_fwd(B, b, B.x[b], B.x[b+1]);
    out_fwd(B, b+1, B.x[b+1]);
  }
  k_mol_sum<<<gr1((long long)NM*128),256,0,stream>>>(B.Pn, B.gmol);
  ggemm(B, B.gmol, B.h0w, B.h0b, B.y0, B.Zh0, NM, 128, 128, 0, 1, 0);
  ggemm(B, B.y0,   B.h1w, B.h1b, B.y1, B.Zh1, NM,  64, 128, 0, 1, 0);
  ggemm(B, B.y1,   B.h2w, B.h2b, B.y2, B.Zh2, NM,  64,  64, 0, 1, 0);
  k_head_last<<<2,64,0,stream>>>(B.y2, B.h3w, B.h3b, preds);

  // ---------------- backward (forces) ----------------
  k_head_binit<<<gr1(NM*64),256,0,stream>>>(B.dh2s, B.h3w);
  k_silu_bwd<<<gr1(NM*64),256,0,stream>>>(B.dh2s, B.dh2s, B.Zh2, NM*64);
  ggemm(B, B.dh2s, B.h2w, nullptr, B.th1, nullptr, NM, 64, 64, 1, 0, 0);
  k_silu_bwd<<<gr1(NM*64),256,0,stream>>>(B.th1, B.th1, B.Zh1, NM*64);
  ggemm(B, B.th1, B.h1w, nullptr, B.th0, nullptr, NM, 128, 64, 1, 0, 0);
  k_silu_bwd<<<gr1(NM*128),256,0,stream>>>(B.th0, B.th0, B.Zh0, NM*128);
  ggemm(B, B.th0, B.h0w, nullptr, B.dgm, nullptr, NM, 128, 128, 1, 0, 0);
  k_node_bcast<<<gr1((long long)NN*128),256,0,stream>>>(B.dPn, B.dgm);

  k_fill<<<gr1((long long)NE*6),256,0,stream>>>(B.drbfA, 0.f, (long long)NE*6);
  k_fill<<<gr1((long long)NE*42),256,0,stream>>>(B.dsrbfA, 0.f, (long long)NE*42);
  k_fill<<<gr1(NT),256,0,stream>>>(B.dcosA, 0.f, NT);
  k_fill<<<gr1(nE),256,0,stream>>>(B.dX, 0.f, nE);

  out_bwd(B, 4, B.x[4], B.dX);
  float* dXcur = B.dX;
  float* dXnew = B.dXn;
  for (int b=3;b>=0;--b){
    k_fill<<<gr1(nE),256,0,stream>>>(dXnew, 0.f, nE);
    int_bwd(B, b, B.x[b], dXcur, dXnew);
    out_bwd(B, b, B.x[b], dXnew);
    float* t = dXcur; dXcur = dXnew; dXnew = t;
  }
  // embedding backward: dx0 -> drbf via rbf_h slice of emb_lin
  k_silu_bwd<<<gr1(nE),256,0,stream>>>(B.v0, dXcur, B.Zemb, nE);
  ggemm(B, B.v0, B.el_w + (size_t)256*128, nullptr, B.v1, nullptr, NE, 128, 128, 1, 0, 0);
  k_silu_bwd<<<gr1(nE),256,0,stream>>>(B.v1, B.v1, B.Zer, nE);
  ggemm(B, B.v1, B.er_w, nullptr, B.drbfA, nullptr, NE, 6, 128, 1, 0, 1);

  // geometry backward -> forces
  k_xs_grad<<<gr1(NE),256,0,stream>>>(B.drbfA, B.rbfd, B.dsrbfA, B.srbfd, B.gdist);
  k_forces<<<gr1(NN),256,0,stream>>>(B.pos, B.esrc, B.edst, B.ikj, B.iji, B.gdist, B.dcosA, forces);
}